// VarianceLoss_26285199851691
// MI455X (gfx1250) — compile-verified
//
#include <hip/hip_runtime.h>

// Problem constants (from reference): S=16 snakes, H=4096 rows, W=1024 cols.
#define S_    16
#define H_    4096
#define W_    1024
#define NSEG  16
#define SEGH  (H_ / NSEG)        // 256 rows per segment
#define TILES_PER_SEG (SEGH/16)  // 16 tiles of 16 rows

typedef __attribute__((ext_vector_type(2))) float v2f;
typedef __attribute__((ext_vector_type(8))) float v8f;

// Workspace layout (float units):
//   [0      , 16384)            seg_sum[s][w]
//   [16384  , 32768)            segoff[s][w]   (exclusive scan of seg_sum over s)
//   [32768  , 33792)            mags[w]
//   [33792  , 33792 + 51*16384) partial[idx][s][w]; idx: 0..16 = A_i, 17..33 = B_i, 34..50 = C_i
#define WS_SEGOFF  16384
#define WS_MAGS    32768
#define WS_PART    33792

// ---------------- Phase A: per-segment column sums (coalesced) ----------------
__global__ void __launch_bounds__(256) k_segsum(const float* __restrict__ preds,
                                                float* __restrict__ ws) {
    int u = blockIdx.x * blockDim.x + threadIdx.x;   // 0..16383
    int s = u >> 10;
    int w = u & (W_ - 1);
    const float* p = preds + (size_t)s * SEGH * W_ + w;
    float acc = 0.f;
#pragma unroll 8
    for (int h = 0; h < SEGH; ++h) acc += p[h * W_];
    ws[u] = acc;
}

// ---------------- Phase B: exclusive scan over segments, mags ----------------
__global__ void __launch_bounds__(256) k_scanseg(float* __restrict__ ws) {
    int w = blockIdx.x * blockDim.x + threadIdx.x;   // 0..1023
    float run = 0.f;
#pragma unroll
    for (int s = 0; s < NSEG; ++s) {
        float v = ws[s * W_ + w];
        ws[WS_SEGOFF + s * W_ + w] = run;
        run += v;
    }
    ws[WS_MAGS + w] = run;  // mags[w] = total column sum
}

// ---------------- Phase C: WMMA tile-scan + sigmoid band accumulation --------
// One wave per (segment, 16-column group). 1024 waves total.
__global__ void __launch_bounds__(128) k_main(const float* __restrict__ snakes,
                                              const float* __restrict__ preds,
                                              float* __restrict__ ws) {
    const int lane = threadIdx.x & 31;
    const int unit = blockIdx.x * (blockDim.x >> 5) + (threadIdx.x >> 5); // 0..1023
    const int g    = unit & 63;    // column group (16 cols each)
    const int s    = unit >> 6;    // segment
    const int m    = lane & 15;    // matrix row (A) / column (B,C,D) within tile
    const int hi   = lane >> 4;    // half-wave select
    const int col  = (g << 4) + m; // this lane's output column

    // Lower-triangular ones matrix L (16x16), sliced into four 16x4 A operands.
    // A layout (f32 16x4): lanes 0-15 M=0..15, VGPR0 = K {0|2}, VGPR1 = K {1|3}.
    v2f aL[4];
#pragma unroll
    for (int j = 0; j < 4; ++j) {
        int k0 = 4 * j + (hi ? 2 : 0);
        int k1 = k0 + 1;
        aL[j][0] = (k0 <= m) ? 1.0f : 0.0f;
        aL[j][1] = (k1 <= m) ? 1.0f : 0.0f;
    }

    // Snake boundaries for this lane's column (both half-waves: same column).
    float snk[16];
#pragma unroll
    for (int i = 0; i < 16; ++i) snk[i] = snakes[i * W_ + col];

    // Incoming cross-segment scan carry for this column.
    float carry = ws[WS_SEGOFF + s * W_ + col];

    float accA[17], accB[17], accC[17];
#pragma unroll
    for (int i = 0; i < 17; ++i) { accA[i] = 0.f; accB[i] = 0.f; accC[i] = 0.f; }

    const float LOG2E = 1.4426950408889634f;

    for (int t = 0; t < TILES_PER_SEG; ++t) {
        const int h0 = s * SEGH + t * 16;
        const float* base = preds + (size_t)h0 * W_ + col;
        if (t + 1 < TILES_PER_SEG)
            __builtin_prefetch(base + 16 * W_, 0, 1);  // global_prefetch_b8

        // C initialized to the running carry, broadcast down each column:
        // D = L*T + carry gives the inclusive column prefix including carry.
        v8f c;
#pragma unroll
        for (int v = 0; v < 8; ++v) c[v] = carry;

        // 16x16x16 scan as 4 chained f32 WMMAs (K=4 each), exact f32.
        // B layout (f32 4x16): VGPR0 = rows {0|2}, VGPR1 = rows {1|3}, N = lane&15.
#pragma unroll
        for (int j = 0; j < 4; ++j) {
            int r0 = 4 * j + (hi ? 2 : 0);
            v2f b;
            b[0] = base[r0 * W_];
            b[1] = base[(r0 + 1) * W_];
            c = __builtin_amdgcn_wmma_f32_16x16x4_f32(
                    /*neg_a=*/false, aL[j], /*neg_b=*/false, b,
                    /*c_mod=*/(short)0, c, /*reuse_a=*/false, /*reuse_b=*/false);
        }

        // New carry = row 15 of D = VGPR7 in lanes 16..31 (column = lane&15).
        carry = __shfl(c[7], 16 + m, 32);

        // Band accumulation: 8 rows per lane live in the D accumulator.
#pragma unroll
        for (int v = 0; v < 8; ++v) {
            const float hf  = (float)(h0 + v + (hi ? 8 : 0));
            const float cun = c[v];          // un-normalized cum at (h, col)
            const float cu2 = cun * cun;

            float sm[17];
            float prev = 0.f, denom = 0.f;
#pragma unroll
            for (int i = 0; i < 16; ++i) {
                // sigmoid(snk - h) = 1 / (1 + exp(h - snk))
                float e   = __builtin_amdgcn_exp2f((hf - snk[i]) * LOG2E);
                float sig = __builtin_amdgcn_rcpf(1.0f + e);
                float d   = sig - prev;
                d = d > 0.f ? d : 0.f;       // clip
                sm[i] = d;  denom += d;  prev = sig;
            }
            {
                float d = 1.0f - prev;
                d = d > 0.f ? d : 0.f;
                sm[16] = d;  denom += d;
            }
            const float rden = __builtin_amdgcn_rcpf(denom);
#pragma unroll
            for (int i = 0; i < 17; ++i) {
                float t2 = sm[i] * rden;     // normalized snakemap
                accA[i] += t2;
                accB[i] = __builtin_fmaf(t2, cun, accB[i]);
                accC[i] = __builtin_fmaf(t2, cu2, accC[i]);
            }
        }
    }

    // Fold the two half-waves (same column) and store per-segment partials.
    float* part = ws + WS_PART;
#pragma unroll
    for (int i = 0; i < 17; ++i) {
        float a  = accA[i] + __shfl_xor(accA[i], 16, 32);
        float b  = accB[i] + __shfl_xor(accB[i], 16, 32);
        float cc = accC[i] + __shfl_xor(accC[i], 16, 32);
        if (lane < 16) {
            part[(size_t)(i)      * (NSEG * W_) + s * W_ + col] = a;
            part[(size_t)(17 + i) * (NSEG * W_) + s * W_ + col] = b;
            part[(size_t)(34 + i) * (NSEG * W_) + s * W_ + col] = cc;
        }
    }
}

// ---------------- Phase D: closed-form variance + masked mean ----------------
__global__ void __launch_bounds__(1024) k_final(const float* __restrict__ ws,
                                                float* __restrict__ out) {
    __shared__ float rsum[1024];
    __shared__ float rcnt[1024];
    const int w = threadIdx.x;
    const float* part = ws + WS_PART;
    const float mags = ws[WS_MAGS + w];

    float acc = 0.f;
    for (int i = 0; i < 17; ++i) {
        float A = 0.f, B = 0.f, C = 0.f;
        for (int s = 0; s < NSEG; ++s) {
            A += part[(size_t)(i)      * (NSEG * W_) + s * W_ + w];
            B += part[(size_t)(17 + i) * (NSEG * W_) + s * W_ + w];
            C += part[(size_t)(34 + i) * (NSEG * W_) + s * W_ + w];
        }
        acc += C - B * B / A;   // sum_h (cum - wmean_i)^2 * sm_i  (un-normalized cum)
    }
    // Undo cum normalization (mags^2) and take mean over (S+1, H).
    const float wstd = acc / (17.0f * (float)H_ * mags * mags);
    const bool mask = mags > 1.0f;
    rsum[w] = mask ? wstd : 0.f;
    rcnt[w] = mask ? 1.f : 0.f;
    __syncthreads();
    for (int off = 512; off > 0; off >>= 1) {
        if (w < off) { rsum[w] += rsum[w + off]; rcnt[w] += rcnt[w + off]; }
        __syncthreads();
    }
    if (w == 0) out[0] = rsum[0] / rcnt[0];
}

extern "C" void kernel_launch(void* const* d_in, const int* in_sizes, int n_in,
                              void* d_out, int out_size, void* d_ws, size_t ws_size,
                              hipStream_t stream) {
    const float* snakes = (const float*)d_in[0];   // [16, 1024]
    const float* preds  = (const float*)d_in[1];   // [4096, 1024]
    float* ws  = (float*)d_ws;
    float* out = (float*)d_out;

    k_segsum <<<64, 256, 0, stream>>>(preds, ws);
    k_scanseg<<<4, 256, 0, stream>>>(ws);
    k_main   <<<256, 128, 0, stream>>>(snakes, preds, ws);
    k_final  <<<1, 1024, 0, stream>>>(ws, out);
}